// InvertAffine_73392401154578
// MI455X (gfx1250) — compile-verified
//
#include <hip/hip_runtime.h>

// ---------------------------------------------------------------------------
// Batched inverse of 4x4 homogeneous affine matrices (top 3x4 rows returned).
//   M = I4 + [trf; 0]  ->  out = [R^-1 | -R^-1 t]   (closed-form adjugate)
//
// MI455X reasoning: 96 B and ~60 FLOPs per matrix -> pure HBM streaming,
// floor ~17us at 23.3 TB/s. Optimization = perfectly coalesced b128 streams
// with minimal per-tile instruction overhead.
// CDNA5 paths used:
//   - GLOBAL_LOAD_ASYNC_TO_LDS_B128  (ASYNCcnt) for coalesced stage-in
//   - GLOBAL_STORE_ASYNC_FROM_LDS_B128 (ASYNCcnt) for coalesced stage-out
//   - s_wait_asynccnt 0 + workgroup barrier for visibility
// Full tiles (the common case: B = 2^22 is a multiple of 256) take an
// unpredicated path: no 64-bit bounds compares, no exec save/restore.
// ---------------------------------------------------------------------------

typedef float v4f __attribute__((ext_vector_type(4)));
typedef int   v4i __attribute__((ext_vector_type(4)));

#define TPB 256

#if defined(__AMDGCN__) && __has_builtin(__builtin_amdgcn_global_load_async_to_lds_b128)
#define USE_ASYNC_LD 1
#else
#define USE_ASYNC_LD 0
#endif

#if defined(__AMDGCN__) && __has_builtin(__builtin_amdgcn_global_store_async_from_lds_b128)
#define USE_ASYNC_ST 1
#else
#define USE_ASYNC_ST 0
#endif

#if defined(__AMDGCN__)
typedef __attribute__((address_space(1))) v4i GI4;  // global (AS1) int4
typedef __attribute__((address_space(3))) v4i LI4;  // LDS   (AS3) int4
#endif

__device__ __forceinline__ void wait_asynccnt0() {
#if defined(__AMDGCN__) && __has_builtin(__builtin_amdgcn_s_wait_asynccnt)
  __builtin_amdgcn_s_wait_asynccnt(0);
#elif defined(__AMDGCN__)
  asm volatile("s_wait_asynccnt 0" ::: "memory");
#endif
}

// Stage-in of one float4, global -> LDS.
__device__ __forceinline__ void load4(const float* __restrict__ g, v4f* l) {
#if USE_ASYNC_LD
  __builtin_amdgcn_global_load_async_to_lds_b128((GI4*)(const v4i*)g, (LI4*)(v4i*)l,
                                                 /*offset=*/0, /*cpol=*/0);
#else
  *l = __builtin_nontemporal_load((const v4f*)g);
#endif
}

// Stage-out of one float4, LDS -> global.
__device__ __forceinline__ void store4(float* __restrict__ g, const v4f* l) {
#if USE_ASYNC_ST
  __builtin_amdgcn_global_store_async_from_lds_b128((GI4*)(v4i*)g, (LI4*)(const v4i*)l,
                                                    /*offset=*/0, /*cpol=*/0);
#else
  __builtin_nontemporal_store(*l, (v4f*)g);
#endif
}

// Closed-form inverse of I3+A with translation, 12 floats in LDS -> 12 out.
__device__ __forceinline__ void invert_one(const v4f* __restrict__ s_in,
                                           v4f* __restrict__ s_out, int t) {
  v4f a = s_in[t * 3 + 0];  // A00 A01 A02 T0
  v4f b = s_in[t * 3 + 1];  // A10 A11 A12 T1
  v4f c = s_in[t * 3 + 2];  // A20 A21 A22 T2

  float r00 = a.x + 1.0f, r01 = a.y,        r02 = a.z,        t0 = a.w;
  float r10 = b.x,        r11 = b.y + 1.0f, r12 = b.z,        t1 = b.w;
  float r20 = c.x,        r21 = c.y,        r22 = c.z + 1.0f, t2 = c.w;

  float c00 = r11 * r22 - r12 * r21;
  float c01 = r12 * r20 - r10 * r22;
  float c02 = r10 * r21 - r11 * r20;
  float det = r00 * c00 + r01 * c01 + r02 * c02;
  float id  = 1.0f / det;

  float i00 = c00 * id;
  float i10 = c01 * id;
  float i20 = c02 * id;
  float i01 = (r02 * r21 - r01 * r22) * id;
  float i11 = (r00 * r22 - r02 * r20) * id;
  float i21 = (r01 * r20 - r00 * r21) * id;
  float i02 = (r01 * r12 - r02 * r11) * id;
  float i12 = (r02 * r10 - r00 * r12) * id;
  float i22 = (r00 * r11 - r01 * r10) * id;

  float o0 = -(i00 * t0 + i01 * t1 + i02 * t2);
  float o1 = -(i10 * t0 + i11 * t1 + i12 * t2);
  float o2 = -(i20 * t0 + i21 * t1 + i22 * t2);

  v4f r0 = {i00, i01, i02, o0};
  v4f r1 = {i10, i11, i12, o1};
  v4f r2 = {i20, i21, i22, o2};
  s_out[t * 3 + 0] = r0;
  s_out[t * 3 + 1] = r1;
  s_out[t * 3 + 2] = r2;
}

__global__ __launch_bounds__(TPB) void InvertAffine_73392401154578_kernel(
    const float* __restrict__ in, float* __restrict__ out, int nmats) {
  __shared__ v4f s_in[TPB * 3];   // 12 KB: staged input tile (256 matrices)
  __shared__ v4f s_out[TPB * 3];  // 12 KB: staged output tile

  const int t = threadIdx.x;
  const long long f4base = (long long)blockIdx.x * (TPB * 3);  // tile start (float4 units)
  const long long total4 = (long long)nmats * 3;
  const bool full = ((long long)(blockIdx.x + 1) * TPB) <= (long long)nmats;

  // ---- Stage in: 3 coalesced async b128 transfers per thread ----
  if (full) {
#pragma unroll
    for (int k = 0; k < 3; ++k) {
      long long i4 = f4base + (long long)(k * TPB + t);
      load4(in + i4 * 4, &s_in[k * TPB + t]);
    }
  } else {
#pragma unroll
    for (int k = 0; k < 3; ++k) {
      long long i4 = f4base + (long long)(k * TPB + t);
      if (i4 < total4) load4(in + i4 * 4, &s_in[k * TPB + t]);
    }
  }
  wait_asynccnt0();
  __syncthreads();

  // ---- Per-thread closed-form inverse (unpredicated on full tiles) ----
  if (full) {
    invert_one(s_in, s_out, t);
  } else if (blockIdx.x * TPB + t < nmats) {
    invert_one(s_in, s_out, t);
  }
  __syncthreads();

  // ---- Stage out: 3 coalesced async b128 transfers per thread ----
  if (full) {
#pragma unroll
    for (int k = 0; k < 3; ++k) {
      long long i4 = f4base + (long long)(k * TPB + t);
      store4(out + i4 * 4, &s_out[k * TPB + t]);
    }
  } else {
#pragma unroll
    for (int k = 0; k < 3; ++k) {
      long long i4 = f4base + (long long)(k * TPB + t);
      if (i4 < total4) store4(out + i4 * 4, &s_out[k * TPB + t]);
    }
  }
  // S_ENDPGM performs an implicit wait-idle (all counters, incl. ASYNCcnt).
}

extern "C" void kernel_launch(void* const* d_in, const int* in_sizes, int n_in,
                              void* d_out, int out_size, void* d_ws, size_t ws_size,
                              hipStream_t stream) {
  const float* trf = (const float*)d_in[0];
  float* out = (float*)d_out;
  const int nmats = in_sizes[0] / 12;  // (B, 3, 4) float32
  const int blocks = (nmats + TPB - 1) / TPB;
  hipLaunchKernelGGL(InvertAffine_73392401154578_kernel, dim3(blocks), dim3(TPB),
                     0, stream, trf, out, nmats);
}